// WeatherMeshEncoder_5299989643913
// MI455X (gfx1250) — compile-verified
//
#include <hip/hip_runtime.h>
#include <hip/hip_bf16.h>
#include <math.h>

// ---------------------------------------------------------------------------
// CDNA5 (gfx1250) WeatherMesh encoder — fp32 WMMA everywhere GEMM-shaped.
// Round 5: qkv/proj GEMMs now stage the shared B (weight) tile in LDS via
// GLOBAL_LOAD_ASYNC_TO_LDS_B128 (ASYNCcnt + s_wait_asynccnt), 4 waves per
// workgroup consume it (macro-tile M=128). Convs keep the round-4 implicit
// GEMM with K-major transposed weights and 32x32 tiles.
// ---------------------------------------------------------------------------

typedef __attribute__((ext_vector_type(2))) float v2f;
typedef __attribute__((ext_vector_type(8))) float v8f;

__device__ __forceinline__ v8f wmma4(v2f a, v2f b, v8f c) {
  // D = A(16x4) * B(4x16) + C, fp32 in / fp32 accum
  return __builtin_amdgcn_wmma_f32_16x16x4_f32(
      /*neg_a=*/false, a, /*neg_b=*/false, b,
      /*c_mod=*/(short)0, c, /*reuse_a=*/false, /*reuse_b=*/false);
}

__device__ __forceinline__ float gelu_tanh(float x) {
  float x3 = x * x * x;
  return 0.5f * x * (1.0f + tanhf(0.7978845608028654f * (x + 0.044715f * x3)));
}

// Async copy 16B from global to LDS (per-lane), tracked by ASYNCcnt.
__device__ __forceinline__ void async_copy_b128(unsigned lds_byte_off,
                                                const void* g) {
  asm volatile("global_load_async_to_lds_b128 %0, %1, off"
               :
               : "v"(lds_byte_off), "v"(g)
               : "memory");
}
__device__ __forceinline__ void wait_async0() {
  asm volatile("s_wait_asynccnt 0" ::: "memory");
}

// ---------------------------------------------------------------------------
// Weight transpose pre-pass: w[Cout][Cin][taps] -> wt[taps][Cin4][Cout],
// zero-padded in ci (Cin4 = ceil4(Cin)).
// ---------------------------------------------------------------------------
__global__ __launch_bounds__(256) void transpose_conv_w(
    const float* __restrict__ w, float* __restrict__ wt, int Cout, int Cin,
    int taps, int Cin4) {
  int idx = blockIdx.x * 256 + threadIdx.x;
  int total = taps * Cin4 * Cout;
  if (idx >= total) return;
  int co = idx % Cout;
  int r = idx / Cout;
  int ci = r % Cin4;
  int t = r / Cin4;
  wt[idx] = (ci < Cin) ? w[(co * Cin + ci) * taps + t] : 0.f;
}

// ---------------------------------------------------------------------------
// conv2d (k=3, stride=2, pad=1) + bias + GELU, implicit GEMM, 32x32 tile.
// x: [Cin, Hin, Win]   wt: [9][Cin4][Cout]   y: [Cout, Hout, Wout]
// ---------------------------------------------------------------------------
template <bool TAIL>
__global__ __launch_bounds__(32) void conv2d_gelu_wmma(
    const float* __restrict__ x, const float* __restrict__ wt,
    const float* __restrict__ bias, float* __restrict__ y, int Cin, int Cout,
    int Hin, int Win) {
  const int Hout = Hin >> 1, Wout = Win >> 1;
  const int Npos = Hout * Wout;
  const int lane = threadIdx.x;
  const int half = lane >> 4, lidx = lane & 15;
  const int n0 = blockIdx.x * 32;   // position tile (x2)
  const int co0 = blockIdx.y * 32;  // cout tile (x2)
  const int CHW = Hin * Win;
  const int Cin4 = (Cin + 3) & ~3;
  const int slab = Cin4 * Cout;  // wt slab per tap

  const int np0 = n0 + lidx, np1 = np0 + 16;
  const int ho0 = np0 / Wout, wo0 = np0 - ho0 * Wout;
  const int ho1 = np1 / Wout, wo1 = np1 - ho1 * Wout;
  const int colA = co0 + lidx;  // lane-contiguous A column (cout)

  v8f c00 = {}, c01 = {}, c10 = {}, c11 = {};  // c[aTile][bTile]
  for (int ty = 0; ty < 3; ++ty) {
    const int ih0 = ho0 * 2 - 1 + ty;
    const int ih1 = ho1 * 2 - 1 + ty;
    for (int tx = 0; tx < 3; ++tx) {
      const int t = ty * 3 + tx;
      const int iw0 = wo0 * 2 - 1 + tx;
      const int iw1 = wo1 * 2 - 1 + tx;
      const bool ok0 =
          ((unsigned)ih0 < (unsigned)Hin) & ((unsigned)iw0 < (unsigned)Win);
      const bool ok1 =
          ((unsigned)ih1 < (unsigned)Hin) & ((unsigned)iw1 < (unsigned)Win);
      const float m0 = ok0 ? 1.f : 0.f;
      const float m1 = ok1 ? 1.f : 0.f;
      const int sp0 = ok0 ? (ih0 * Win + iw0) : 0;  // clamped, in-bounds
      const int sp1 = ok1 ? (ih1 * Win + iw1) : 0;
      const float* wtt = wt + t * slab;
      int offA = (2 * half) * Cout + colA;  // walks by 4*Cout
      int offB0 = sp0 + (2 * half) * CHW;   // walks by 4*CHW
      int offB1 = sp1 + (2 * half) * CHW;
#pragma unroll 2
      for (int ci = 0; ci < Cin4; ci += 4) {
        v2f a0, a1, b0, b1;
        a0.x = wtt[offA];
        a0.y = wtt[offA + Cout];
        a1.x = wtt[offA + 16];
        a1.y = wtt[offA + Cout + 16];
        offA += 4 * Cout;
        if (TAIL) {  // A rows zero-padded: B only needs address clamp
          const int k0 = ci + 2 * half, k1 = k0 + 1;
          const int cc0 = min(k0, Cin - 1), cc1 = min(k1, Cin - 1);
          b0.x = x[sp0 + cc0 * CHW] * m0;
          b0.y = x[sp0 + cc1 * CHW] * m0;
          b1.x = x[sp1 + cc0 * CHW] * m1;
          b1.y = x[sp1 + cc1 * CHW] * m1;
        } else {
          b0.x = x[offB0] * m0;
          b0.y = x[offB0 + CHW] * m0;
          b1.x = x[offB1] * m1;
          b1.y = x[offB1 + CHW] * m1;
          offB0 += 4 * CHW;
          offB1 += 4 * CHW;
        }
        c00 = wmma4(a0, b0, c00);
        c10 = wmma4(a1, b0, c10);
        c01 = wmma4(a0, b1, c01);
        c11 = wmma4(a1, b1, c11);
      }
    }
  }
#pragma unroll
  for (int j = 0; j < 8; ++j) {
    const int m = j + 8 * half;
    const int r0 = co0 + m, r1 = co0 + 16 + m;
    const float bb0 = bias[r0], bb1 = bias[r1];
    y[r0 * Npos + n0 + lidx] = gelu_tanh(c00[j] + bb0);
    y[r1 * Npos + n0 + lidx] = gelu_tanh(c10[j] + bb1);
    y[r0 * Npos + n0 + 16 + lidx] = gelu_tanh(c01[j] + bb0);
    y[r1 * Npos + n0 + 16 + lidx] = gelu_tanh(c11[j] + bb1);
  }
}

// ---------------------------------------------------------------------------
// conv3d (k=3x3x3, stride=(1,2,2), pad=1) + bias + GELU, 32x32 tile.
// x: [Cin, Din, Hin, Win]  wt: [27][Cin4][Cout]  y: [Cout, Din, Hout, Wout]
// ---------------------------------------------------------------------------
template <bool TAIL>
__global__ __launch_bounds__(32) void conv3d_gelu_wmma(
    const float* __restrict__ x, const float* __restrict__ wt,
    const float* __restrict__ bias, float* __restrict__ y, int Cin, int Cout,
    int Din, int Hin, int Win) {
  const int Hout = Hin >> 1, Wout = Win >> 1;
  const int HWo = Hout * Wout;
  const int Npos = Din * HWo;
  const int lane = threadIdx.x;
  const int half = lane >> 4, lidx = lane & 15;
  const int n0 = blockIdx.x * 32;
  const int co0 = blockIdx.y * 32;
  const int CDHW = Din * Hin * Win;
  const int Cin4 = (Cin + 3) & ~3;
  const int slab = Cin4 * Cout;

  const int np0 = n0 + lidx, np1 = np0 + 16;
  const int dd0 = np0 / HWo, q0 = np0 - dd0 * HWo;
  const int ho0 = q0 / Wout, wo0 = q0 - ho0 * Wout;
  const int dd1 = np1 / HWo, q1 = np1 - dd1 * HWo;
  const int ho1 = q1 / Wout, wo1 = q1 - ho1 * Wout;
  const int colA = co0 + lidx;

  v8f c00 = {}, c01 = {}, c10 = {}, c11 = {};
  for (int tz = 0; tz < 3; ++tz) {
    const int id0 = dd0 - 1 + tz;
    const int id1 = dd1 - 1 + tz;
    const bool okd0 = (unsigned)id0 < (unsigned)Din;
    const bool okd1 = (unsigned)id1 < (unsigned)Din;
    for (int ty = 0; ty < 3; ++ty) {
      const int ih0 = ho0 * 2 - 1 + ty;
      const int ih1 = ho1 * 2 - 1 + ty;
      const bool okh0 = okd0 & ((unsigned)ih0 < (unsigned)Hin);
      const bool okh1 = okd1 & ((unsigned)ih1 < (unsigned)Hin);
      for (int tx = 0; tx < 3; ++tx) {
        const int t = (tz * 3 + ty) * 3 + tx;
        const int iw0 = wo0 * 2 - 1 + tx;
        const int iw1 = wo1 * 2 - 1 + tx;
        const bool ok0 = okh0 & ((unsigned)iw0 < (unsigned)Win);
        const bool ok1 = okh1 & ((unsigned)iw1 < (unsigned)Win);
        const float m0 = ok0 ? 1.f : 0.f;
        const float m1 = ok1 ? 1.f : 0.f;
        const int sp0 = ok0 ? ((id0 * Hin + ih0) * Win + iw0) : 0;
        const int sp1 = ok1 ? ((id1 * Hin + ih1) * Win + iw1) : 0;
        const float* wtt = wt + t * slab;
        int offA = (2 * half) * Cout + colA;
        int offB0 = sp0 + (2 * half) * CDHW;
        int offB1 = sp1 + (2 * half) * CDHW;
#pragma unroll 2
        for (int ci = 0; ci < Cin4; ci += 4) {
          v2f a0, a1, b0, b1;
          a0.x = wtt[offA];
          a0.y = wtt[offA + Cout];
          a1.x = wtt[offA + 16];
          a1.y = wtt[offA + Cout + 16];
          offA += 4 * Cout;
          if (TAIL) {
            const int k0 = ci + 2 * half, k1 = k0 + 1;
            const int cc0 = min(k0, Cin - 1), cc1 = min(k1, Cin - 1);
            b0.x = x[sp0 + cc0 * CDHW] * m0;
            b0.y = x[sp0 + cc1 * CDHW] * m0;
            b1.x = x[sp1 + cc0 * CDHW] * m1;
            b1.y = x[sp1 + cc1 * CDHW] * m1;
          } else {
            b0.x = x[offB0] * m0;
            b0.y = x[offB0 + CDHW] * m0;
            b1.x = x[offB1] * m1;
            b1.y = x[offB1 + CDHW] * m1;
            offB0 += 4 * CDHW;
            offB1 += 4 * CDHW;
          }
          c00 = wmma4(a0, b0, c00);
          c10 = wmma4(a1, b0, c10);
          c01 = wmma4(a0, b1, c01);
          c11 = wmma4(a1, b1, c11);
        }
      }
    }
  }
#pragma unroll
  for (int j = 0; j < 8; ++j) {
    const int m = j + 8 * half;
    const int r0 = co0 + m, r1 = co0 + 16 + m;
    const float bb0 = bias[r0], bb1 = bias[r1];
    y[r0 * Npos + n0 + lidx] = gelu_tanh(c00[j] + bb0);
    y[r1 * Npos + n0 + lidx] = gelu_tanh(c10[j] + bb1);
    y[r0 * Npos + n0 + 16 + lidx] = gelu_tanh(c01[j] + bb0);
    y[r1 * Npos + n0 + 16 + lidx] = gelu_tanh(c11[j] + bb1);
  }
}

// ---------------------------------------------------------------------------
// lat = 1x1x1 conv over concat([p2, s2[:,None]], depth), fused gather.
// p2: [1024, 5, 16, 16]  s2: [1024, 16, 16]  wl: [256, 1024]  bl: [256]
// lat: [1536 pos][256]. 32x32 tile.
// ---------------------------------------------------------------------------
__global__ __launch_bounds__(32) void latconv_wmma(
    const float* __restrict__ p2, const float* __restrict__ s2,
    const float* __restrict__ wl, const float* __restrict__ bl,
    float* __restrict__ lat) {
  const int lane = threadIdx.x;
  const int half = lane >> 4, lidx = lane & 15;
  const int m0 = blockIdx.x * 32;  // pos tile (x2)
  const int n0 = blockIdx.y * 32;  // cout tile (x2)

  const int p0 = m0 + lidx, p1 = p0 + 16;
  const int d0 = p0 >> 8, r0 = p0 & 255;
  const int d1 = p1 >> 8, r1 = p1 & 255;
  const float* ab0 = (d0 < 5) ? (p2 + d0 * 256 + r0) : (s2 + r0);
  const float* ab1 = (d1 < 5) ? (p2 + d1 * 256 + r1) : (s2 + r1);
  const int as0 = (d0 < 5) ? 1280 : 256;
  const int as1 = (d1 < 5) ? 1280 : 256;
  const float* bcol0 = wl + (n0 + lidx) * 1024;
  const float* bcol1 = bcol0 + 16 * 1024;

  int offA0 = (2 * half) * as0;
  int offA1 = (2 * half) * as1;
  v8f c00 = {}, c01 = {}, c10 = {}, c11 = {};  // c[aTile(pos)][bTile(cout)]
#pragma unroll 2
  for (int k = 0; k < 1024; k += 4) {
    const int ka = k + 2 * half;
    v2f a0, a1;
    a0.x = ab0[offA0];
    a0.y = ab0[offA0 + as0];
    offA0 += 4 * as0;
    a1.x = ab1[offA1];
    a1.y = ab1[offA1 + as1];
    offA1 += 4 * as1;
    v2f b0 = *(const v2f*)(bcol0 + ka);  // 8B aligned
    v2f b1 = *(const v2f*)(bcol1 + ka);
    c00 = wmma4(a0, b0, c00);
    c01 = wmma4(a0, b1, c01);
    c10 = wmma4(a1, b0, c10);
    c11 = wmma4(a1, b1, c11);
  }
  const float bb0 = bl[n0 + lidx], bb1 = bl[n0 + 16 + lidx];
#pragma unroll
  for (int j = 0; j < 8; ++j) {
    const int ma = m0 + j + 8 * half;
    lat[ma * 256 + n0 + lidx] = c00[j] + bb0;
    lat[ma * 256 + n0 + 16 + lidx] = c01[j] + bb1;
    lat[(ma + 16) * 256 + n0 + lidx] = c10[j] + bb0;
    lat[(ma + 16) * 256 + n0 + 16 + lidx] = c11[j] + bb1;
  }
}

// ---------------------------------------------------------------------------
// GEMM + bias with async-LDS staged B tile.
// C[M][N] = A[M][K] * B[K][N] + b.  K <= 256 (qkv/proj both have K=256).
// Workgroup = 4 waves, macro-tile M=128 x N=32; the 32KB B tile (K x 32) is
// staged once into LDS via GLOBAL_LOAD_ASYNC_TO_LDS_B128 (ASYNCcnt), then
// all 4 waves consume it from LDS (ds_load) while A streams from global.
// ---------------------------------------------------------------------------
__global__ __launch_bounds__(128) void gemm_bias_lds_wmma(
    const float* __restrict__ A, const float* __restrict__ B,
    const float* __restrict__ bias, float* __restrict__ C, int N, int K) {
  __shared__ float Bs[256 * 32];  // K-major B tile, 32KB
  const int tid = threadIdx.x;
  const int wave = tid >> 5;
  const int lane = tid & 31;
  const int half = lane >> 4, lidx = lane & 15;
  const int m0 = blockIdx.x * 128 + wave * 32;  // per-wave M tile (x2)
  const int n0 = blockIdx.y * 32;               // N tile (x2)

  // ---- cooperative async stage of B[k][n0..n0+32), k = 0..K ----
  {
    const int nchunks = K * 8;  // 16B chunks (4 floats), 8 per row
    for (int c = tid; c < nchunks; c += 128) {
      const int k = c >> 3;
      const int cc = (c & 7) * 4;
      const unsigned ldsoff = (unsigned)(size_t)(&Bs[k * 32 + cc]);
      async_copy_b128(ldsoff, B + k * N + n0 + cc);
    }
    wait_async0();       // ASYNCcnt == 0: this wave's copies landed
    __syncthreads();     // all waves' copies landed
  }

  const float* arow0 = A + (m0 + lidx) * K;  // contiguous in K
  const float* arow1 = arow0 + 16 * K;

  v8f c00 = {}, c01 = {}, c10 = {}, c11 = {};  // c[aTile(M)][bTile(N)]
#pragma unroll 4
  for (int k = 0; k < K; k += 4) {
    const int ka = k + 2 * half;
    v2f a0 = *(const v2f*)(arow0 + ka);  // 8B aligned
    v2f a1 = *(const v2f*)(arow1 + ka);
    v2f b0, b1;
    b0.x = Bs[ka * 32 + lidx];
    b0.y = Bs[(ka + 1) * 32 + lidx];
    b1.x = Bs[ka * 32 + 16 + lidx];
    b1.y = Bs[(ka + 1) * 32 + 16 + lidx];
    c00 = wmma4(a0, b0, c00);
    c10 = wmma4(a1, b0, c10);
    c01 = wmma4(a0, b1, c01);
    c11 = wmma4(a1, b1, c11);
  }
  const float bb0 = bias[n0 + lidx], bb1 = bias[n0 + 16 + lidx];
#pragma unroll
  for (int j = 0; j < 8; ++j) {
    const int m = m0 + j + 8 * half;
    C[m * N + n0 + lidx] = c00[j] + bb0;
    C[m * N + n0 + 16 + lidx] = c01[j] + bb1;
    C[(m + 16) * N + n0 + lidx] = c10[j] + bb0;
    C[(m + 16) * N + n0 + 16 + lidx] = c11[j] + bb1;
  }
}

// ---------------------------------------------------------------------------
// NATTEN 3D neighborhood attention, window 5x7x7 over D=6,H=16,W=16, hd=32.
// One thread per (pos, head); online (flash) softmax over 245 neighbors.
// qkv: [1536][768] (q|k|v)   att: [1536][256]
// ---------------------------------------------------------------------------
__global__ __launch_bounds__(256) void natten_kernel(
    const float* __restrict__ qkv, float* __restrict__ att) {
  int t = blockIdx.x * blockDim.x + threadIdx.x;
  if (t >= 1536 * 8) return;
  int pos = t >> 3, head = t & 7;
  int d = pos >> 8;
  int rem = pos & 255;
  int h = rem >> 4, w = rem & 15;
  int sd = min(max(d - 2, 0), 6 - 5);
  int sh = min(max(h - 3, 0), 16 - 7);
  int sw = min(max(w - 3, 0), 16 - 7);

  const float scale = 0.17677669529663687f;  // 32^-0.5
  float q[32];
  const float* qp = qkv + pos * 768 + head * 32;
#pragma unroll
  for (int e = 0; e < 32; ++e) q[e] = qp[e] * scale;

  float mval = -3.0e38f, ssum = 0.f;
  float acc[32];
#pragma unroll
  for (int e = 0; e < 32; ++e) acc[e] = 0.f;

  for (int dz = 0; dz < 5; ++dz) {
    for (int dy = 0; dy < 7; ++dy) {
      for (int dx = 0; dx < 7; ++dx) {
        int npos = (sd + dz) * 256 + (sh + dy) * 16 + (sw + dx);
        const float* kp = qkv + npos * 768 + 256 + head * 32;
        const float* vp = kp + 256;
        float dot = 0.f;
#pragma unroll
        for (int e = 0; e < 32; ++e) dot = fmaf(q[e], kp[e], dot);
        float nm = fmaxf(mval, dot);
        float corr = __expf(mval - nm);
        float p = __expf(dot - nm);
        ssum = ssum * corr + p;
#pragma unroll
        for (int e = 0; e < 32; ++e) acc[e] = fmaf(p, vp[e], acc[e] * corr);
        mval = nm;
      }
    }
  }
  float inv = 1.f / ssum;
  float* op = att + pos * 256 + head * 32;
#pragma unroll
  for (int e = 0; e < 32; ++e) op[e] = acc[e] * inv;
}

// ---------------------------------------------------------------------------
// Host-side orchestration.
// Input order (setup_inputs insertion order):
//  0: surface (1,8,128,128)   1: pressure (1,5,5,128,128)
//  2..13: w2_i,b2_i,w3_i,b3_i for i=0..2 (interleaved per i)
//  14: wl  15: bl
//  16+4i: wqkv_i  17+4i: bqkv_i  18+4i: wproj_i  19+4i: bproj_i
// ---------------------------------------------------------------------------
static inline void xpose(const float* w, float* wt, int Cout, int Cin, int taps,
                         hipStream_t stream) {
  int Cin4 = (Cin + 3) & ~3;
  int total = taps * Cin4 * Cout;
  transpose_conv_w<<<(total + 255) / 256, 256, 0, stream>>>(w, wt, Cout, Cin,
                                                            taps, Cin4);
}

extern "C" void kernel_launch(void* const* d_in, const int* in_sizes, int n_in,
                              void* d_out, int out_size, void* d_ws,
                              size_t ws_size, hipStream_t stream) {
  (void)in_sizes; (void)n_in; (void)out_size; (void)ws_size;
  const float* surface = (const float*)d_in[0];
  const float* pressure = (const float*)d_in[1];
  const float* w2[3] = {(const float*)d_in[2], (const float*)d_in[6],
                        (const float*)d_in[10]};
  const float* b2[3] = {(const float*)d_in[3], (const float*)d_in[7],
                        (const float*)d_in[11]};
  const float* w3[3] = {(const float*)d_in[4], (const float*)d_in[8],
                        (const float*)d_in[12]};
  const float* b3[3] = {(const float*)d_in[5], (const float*)d_in[9],
                        (const float*)d_in[13]};
  const float* wl = (const float*)d_in[14];
  const float* bl = (const float*)d_in[15];

  const size_t MB = 1024u * 1024u;
  char* ws = (char*)d_ws;
  float* S0 = (float*)(ws + 0);        //  4 MB: s0 (256x64x64), later s2
  float* P0 = (float*)(ws + 4 * MB);   // 20 MB: p0 (256x5x64x64), later p2
  float* S1 = (float*)(ws + 24 * MB);  //  2 MB: s1 (512x32x32), later lat0
  float* P1 = (float*)(ws + 26 * MB);  // 10 MB: p1, later lat1/qkv/att
  float* WT = (float*)(ws + 36 * MB);  // 54 MB: transposed weights (reused)
  float* S2 = S0;
  float* P2 = P0;
  float* LAT0 = S1;
  float* LAT1 = P1;
  float* QKV = (float*)(ws + 26 * MB + 2 * MB);
  float* ATT = (float*)(ws + 26 * MB + 8 * MB);

  // ---- conv tower (surface 2D branch + pressure 3D branch) ----
  xpose(w2[0], WT, 256, 8, 9, stream);
  conv2d_gelu_wmma<false><<<dim3(4096 / 32, 256 / 32), 32, 0, stream>>>(
      surface, WT, b2[0], S0, 8, 256, 128, 128);
  xpose(w3[0], WT, 256, 5, 27, stream);
  conv3d_gelu_wmma<true><<<dim3(20480 / 32, 256 / 32), 32, 0, stream>>>(
      pressure, WT, b3[0], P0, 5, 256, 5, 128, 128);

  xpose(w2[1], WT, 512, 256, 9, stream);
  conv2d_gelu_wmma<false><<<dim3(1024 / 32, 512 / 32), 32, 0, stream>>>(
      S0, WT, b2[1], S1, 256, 512, 64, 64);
  xpose(w3[1], WT, 512, 256, 27, stream);
  conv3d_gelu_wmma<false><<<dim3(5120 / 32, 512 / 32), 32, 0, stream>>>(
      P0, WT, b3[1], P1, 256, 512, 5, 64, 64);

  xpose(w2[2], WT, 1024, 512, 9, stream);
  conv2d_gelu_wmma<false><<<dim3(256 / 32, 1024 / 32), 32, 0, stream>>>(
      S1, WT, b2[2], S2, 512, 1024, 32, 32);
  xpose(w3[2], WT, 1024, 512, 27, stream);
  conv3d_gelu_wmma<false><<<dim3(1280 / 32, 1024 / 32), 32, 0, stream>>>(
      P1, WT, b3[2], P2, 512, 1024, 5, 32, 32);

  // ---- lat conv (concat fused into gather), pos-major output ----
  latconv_wmma<<<dim3(1536 / 32, 256 / 32), 32, 0, stream>>>(P2, S2, wl, bl,
                                                             LAT0);

  // ---- 3 NATTEN layers ----
  float* bufs[2] = {LAT0, LAT1};
  for (int i = 0; i < 3; ++i) {
    const float* wqkv = (const float*)d_in[16 + 4 * i];
    const float* bqkv = (const float*)d_in[17 + 4 * i];
    const float* wproj = (const float*)d_in[18 + 4 * i];
    const float* bproj = (const float*)d_in[19 + 4 * i];
    const float* xin = bufs[i & 1];
    float* xout = (i == 2) ? (float*)d_out : bufs[(i + 1) & 1];

    gemm_bias_lds_wmma<<<dim3(1536 / 128, 768 / 32), 128, 0, stream>>>(
        xin, wqkv, bqkv, QKV, 768, 256);
    natten_kernel<<<dim3(48), 256, 0, stream>>>(QKV, ATT);
    gemm_bias_lds_wmma<<<dim3(1536 / 128, 256 / 32), 128, 0, stream>>>(
        ATT, wproj, bproj, xout, 256, 256);
  }
}